// CTNNBackflowStyleJastrow_75067438399628
// MI455X (gfx1250) — compile-verified
//
#include <hip/hip_runtime.h>
#include <math.h>

// Problem constants (from reference)
#define NB   1024   // batch
#define NN   32     // particles
#define ND   3
#define NH   32
#define NM   32
#define NRH  64
#define NPAIR 496
#define OMEGA_SQRT 1.0f   // OMEGA = 1.0

typedef __attribute__((ext_vector_type(2))) float v2f;
typedef __attribute__((ext_vector_type(8))) float v8f;

__device__ __forceinline__ v8f wmma4(v2f a, v2f b, v8f c) {
  // D = A(16x4,f32) * B(4x16,f32) + C(16x16,f32)
  return __builtin_amdgcn_wmma_f32_16x16x4_f32(
      false, a, false, b, (short)0, c, false, false);
}

__device__ __forceinline__ void wbar() { __builtin_amdgcn_wave_barrier(); }

__device__ __forceinline__ float siluf(float t) { return t / (1.f + __expf(-t)); }

// 16xK A-tile (row-major, stride lda) times W^T where W is [32][K] row-major.
// Produces full 32 output columns: c0 = cols 0..15, c1 = cols 16..31.
// Fragment layouts per CDNA5 ISA 7.12.2 (f32, 16x16x4).
__device__ __forceinline__ void gemm_tile(const float* A, int lda,
                                          const float* W, int K,
                                          v8f& c0, v8f& c1, int lane) {
  const int row  = lane & 15;
  const int koff = (lane >> 4) << 1;   // 0 for lanes 0-15, 2 for lanes 16-31
  const int ncol = lane & 15;
#pragma unroll
  for (int k0 = 0; k0 < K; k0 += 4) {
    const int ka = k0 + koff, kb = ka + 1;
    v2f a;  a.x  = A[row * lda + ka];        a.y  = A[row * lda + kb];
    v2f b0; b0.x = W[ncol * K + ka];         b0.y = W[ncol * K + kb];
    v2f b1; b1.x = W[(16 + ncol) * K + ka];  b1.y = W[(16 + ncol) * K + kb];
    c0 = wmma4(a, b0, c0);
    c1 = wmma4(a, b1, c1);
  }
}

__device__ __forceinline__ void store_frag(float* Sdst, v8f c, int n0, int lane) {
  const int off = (lane & 16) ? 8 : 0;
  const int col = n0 + (lane & 15);
#pragma unroll
  for (int v = 0; v < 8; ++v) Sdst[(v + off) * 32 + col] = c[v];
}

__device__ __forceinline__ v8f addb(v8f c, const float* bias, int n0, int lane) {
  const float bv = bias[n0 + (lane & 15)];
#pragma unroll
  for (int v = 0; v < 8; ++v) c[v] += bv;
  return c;
}

__device__ __forceinline__ v8f addb_silu(v8f c, const float* bias, int n0, int lane) {
  const float bv = bias[n0 + (lane & 15)];
#pragma unroll
  for (int v = 0; v < 8; ++v) { float t = c[v] + bv; c[v] = siluf(t); }
  return c;
}

__global__ __launch_bounds__(128) void ctnn_jastrow_kernel(
    const float* __restrict__ x,
    const float* __restrict__ W_node, const float* __restrict__ b_node,
    const float* __restrict__ We1,    const float* __restrict__ be1,
    const float* __restrict__ We2,    const float* __restrict__ be2,
    const float* __restrict__ Wv2e,   const float* __restrict__ We2v,
    const float* __restrict__ Wu1,    const float* __restrict__ bu1,
    const float* __restrict__ Wu2,    const float* __restrict__ bu2,
    const float* __restrict__ Wn1,    const float* __restrict__ bn1,
    const float* __restrict__ Wn2,    const float* __restrict__ bn2,
    const float* __restrict__ Wf1,    const float* __restrict__ bf1,
    const float* __restrict__ Wf2,    const float* __restrict__ bf2,
    const float* __restrict__ Wf3,    const float* __restrict__ bf3,
    float* __restrict__ out) {

  const int b    = blockIdx.x;
  const int tid  = threadIdx.x;
  const int wave = tid >> 5;
  const int lane = tid & 31;

  // ---- LDS layout (~62.6 KB) ----
  __shared__ float sWe1p[32 * 8];     // We1 padded K=5 -> 8
  __shared__ float sWe2[32 * 32];
  __shared__ float sWu1[32 * 96];
  __shared__ float sWu2[32 * 32];
  __shared__ float sWe2v[32 * 32];
  __shared__ float sbe1[32], sbe2[32], sbu1[32], sbu2[32];
  __shared__ float sx[NN * ND];       // x_sc
  __shared__ float shv[NN * NH];      // h_v (pre-update, kept)
  __shared__ float sp[NN * NM];       // p = h_v @ Wv2e^T
  __shared__ float smv[NN * NH];      // m_v accumulator
  __shared__ float st1[NN * NH];      // node-update hidden
  __shared__ float sheSum[32];        // sum over upper-tri pairs of (h_e + h_e_new)
  __shared__ float shvSum[32];        // sum over nodes of updated h_v
  __shared__ float sfin[130];
  __shared__ float sl1[64], sl2[64];
  __shared__ float sstat[3];          // r2sum, s1sum, cusp
  __shared__ float sEA[4][16 * 8];    // per-wave edge-input tile (padded)
  __shared__ float sS1[4][16 * 32];   // per-wave scratch A
  __shared__ float sS2[4][16 * 32];   // per-wave h_e tile

  const float* xb = x + (size_t)b * NN * ND;

  // ---- Phase A0: zero accumulators ----
  for (int t = tid; t < NN * NH; t += 128) smv[t] = 0.f;
  for (int t = tid; t < 32; t += 128) { sheSum[t] = 0.f; shvSum[t] = 0.f; }
  if (tid < 3) sstat[tid] = 0.f;
  __syncthreads();

  // ---- Phase A1: weights -> LDS, x -> LDS (+ r2 stat), pair stats ----
  for (int t = tid; t < 32 * 8; t += 128) {
    int n = t >> 3, k = t & 7;
    sWe1p[t] = (k < 5) ? We1[n * 5 + k] : 0.f;
  }
  for (int t = tid; t < 1024; t += 128) sWe2[t]  = We2[t];
  for (int t = tid; t < 3072; t += 128) sWu1[t]  = Wu1[t];
  for (int t = tid; t < 1024; t += 128) sWu2[t]  = Wu2[t];
  for (int t = tid; t < 1024; t += 128) sWe2v[t] = We2v[t];
  if (tid < 32) { sbe1[tid] = be1[tid]; sbe2[tid] = be2[tid];
                  sbu1[tid] = bu1[tid]; sbu2[tid] = bu2[tid]; }
  for (int t = tid; t < NN * ND; t += 128) {
    float v = xb[t] * OMEGA_SQRT;
    sx[t] = v;
    atomicAdd(&sstat[0], v * v);               // sum(x_sc^2)
  }
  // pair stats over upper triangle (raw x)
  for (int pr = tid; pr < NPAIR; pr += 128) {
    int i2 = 0, rem = pr;
    while (rem >= (31 - i2)) { rem -= (31 - i2); ++i2; }
    int j2 = i2 + 1 + rem;
    float dx = xb[i2 * 3 + 0] - xb[j2 * 3 + 0];
    float dy = xb[i2 * 3 + 1] - xb[j2 * 3 + 1];
    float dz = xb[i2 * 3 + 2] - xb[j2 * 3 + 2];
    float d2 = dx * dx + dy * dy + dz * dz;
    float dist = sqrtf(d2 + 1e-8f);
    float ds = dist * 5.0f;                    // dist / 0.2
    atomicAdd(&sstat[1], log1pf(ds * ds));
    float rc = sqrtf(d2 + 1e-30f);
    bool same = (i2 < 16) == (j2 < 16);
    float gamma = same ? 0.25f : 0.5f;         // 1/(D+1) : 1/(D-1)
    atomicAdd(&sstat[2], gamma * rc * __expf(-rc));
  }
  __syncthreads();

  // ---- Phase A2: h_v = [x_sc, spin] @ W_node^T + b_node ----
  for (int idx = tid; idx < NN * NH; idx += 128) {
    int node = idx >> 5, h = idx & 31;
    float sf = (node >= 16) ? 1.f : 0.f;
    float a = b_node[h];
    a += sx[node * 3 + 0] * W_node[h * 4 + 0];
    a += sx[node * 3 + 1] * W_node[h * 4 + 1];
    a += sx[node * 3 + 2] * W_node[h * 4 + 2];
    a += sf * W_node[h * 4 + 3];
    shv[idx] = a;
  }
  __syncthreads();

  // ---- Phase A3: p = h_v @ Wv2e^T  (WMMA, waves 0-1) ----
  if (wave < 2) {
    const int m0 = wave << 4;
    v8f c0 = {}, c1 = {};
    gemm_tile(shv + m0 * 32, 32, Wv2e, 32, c0, c1, lane);
    store_frag(sp + m0 * 32, c0, 0, lane);
    store_frag(sp + m0 * 32, c1, 16, lane);
  }
  __syncthreads();

  // ---- Phase B: edge pipeline, 64 tiles of 16 pairs; fixed i per tile ----
  float* EAw = sEA[wave];
  float* S1w = sS1[wave];
  float* S2w = sS2[wave];
  for (int tile = wave; tile < 64; tile += 4) {
    const int i  = tile >> 1;
    const int j0 = (tile & 1) << 4;

    // 1) edge_in tile (16 x 5, padded to 8)
    if (lane < 16) {
      const int j = j0 + lane;
      float rx = sx[i * 3 + 0] - sx[j * 3 + 0];
      float ry = sx[i * 3 + 1] - sx[j * 3 + 1];
      float rz = sx[i * 3 + 2] - sx[j * 3 + 2];
      float r2 = rx * rx + ry * ry + rz * rz;
      float r1 = sqrtf(r2 + 1e-12f);
      float* e = &EAw[lane * 8];
      e[0] = rx; e[1] = ry; e[2] = rz; e[3] = r1; e[4] = r2;
      e[5] = 0.f; e[6] = 0.f; e[7] = 0.f;
    }
    wbar();

    // 2) H1 = silu(edge_in @ We1^T + be1)
    v8f c0 = {}, c1 = {};
    gemm_tile(EAw, 8, sWe1p, 8, c0, c1, lane);
    c0 = addb_silu(c0, sbe1, 0, lane);
    c1 = addb_silu(c1, sbe1, 16, lane);
    wbar();
    store_frag(S1w, c0, 0, lane);
    store_frag(S1w, c1, 16, lane);
    wbar();

    // 3) h_e = H1 @ We2^T + be2
    v8f e0 = {}, e1 = {};
    gemm_tile(S1w, 32, sWe2, 32, e0, e1, lane);
    e0 = addb(e0, sbe2, 0, lane);
    e1 = addb(e1, sbe2, 16, lane);
    wbar();
    store_frag(S2w, e0, 0, lane);
    store_frag(S2w, e1, 16, lane);
    wbar();

    // 4) U1 = silu(eu_in @ Wu1^T + bu1), eu_in = [h_e, p_i, p_j], K=96
    v8f u0 = {}, u1 = {};
    {
      const int row  = lane & 15;
      const int koff = (lane >> 4) << 1;
      const int ncol = lane & 15;
#pragma unroll
      for (int k0 = 0; k0 < 96; k0 += 4) {
        const int ka = k0 + koff, kb = ka + 1;
        v2f a;
        a.x = (ka < 32) ? S2w[row * 32 + ka]
            : (ka < 64) ? sp[i * 32 + (ka - 32)]
                        : sp[(j0 + row) * 32 + (ka - 64)];
        a.y = (kb < 32) ? S2w[row * 32 + kb]
            : (kb < 64) ? sp[i * 32 + (kb - 32)]
                        : sp[(j0 + row) * 32 + (kb - 64)];
        v2f b0; b0.x = sWu1[ncol * 96 + ka];        b0.y = sWu1[ncol * 96 + kb];
        v2f b1; b1.x = sWu1[(16 + ncol) * 96 + ka]; b1.y = sWu1[(16 + ncol) * 96 + kb];
        u0 = wmma4(a, b0, u0);
        u1 = wmma4(a, b1, u1);
      }
    }
    u0 = addb_silu(u0, sbu1, 0, lane);
    u1 = addb_silu(u1, sbu1, 16, lane);
    wbar();
    store_frag(S1w, u0, 0, lane);
    store_frag(S1w, u1, 16, lane);
    wbar();

    // 5) h_e_new = U1 @ Wu2^T + bu2
    v8f hn0 = {}, hn1 = {};
    gemm_tile(S1w, 32, sWu2, 32, hn0, hn1, lane);
    hn0 = addb(hn0, sbu2, 0, lane);
    hn1 = addb(hn1, sbu2, 16, lane);
    wbar();
    store_frag(S1w, hn0, 0, lane);   // S1 now holds h_e_new (A for msgs)
    store_frag(S1w, hn1, 16, lane);
    wbar();

    // 6) accumulate upper-triangle feature: sum over pairs (j>i) of h_e + h_e_new
    {
      const int off = (lane & 16) ? 8 : 0;
      const int cb  = lane & 15;
      float s0 = 0.f, s1 = 0.f;
#pragma unroll
      for (int v = 0; v < 8; ++v) {
        const int row = off + v;
        if ((j0 + row) > i) {
          s0 += S2w[row * 32 + cb]      + hn0[v];
          s1 += S2w[row * 32 + 16 + cb] + hn1[v];
        }
      }
      s0 += __shfl_xor(s0, 16, 32);
      s1 += __shfl_xor(s1, 16, 32);
      if (lane < 16) { atomicAdd(&sheSum[cb], s0); atomicAdd(&sheSum[16 + cb], s1); }
    }

    // 7) msgs = h_e_new @ We2v^T ; m_v[i] += sum_{j != i}
    v8f g0 = {}, g1 = {};
    gemm_tile(S1w, 32, sWe2v, 32, g0, g1, lane);
    {
      const int off = (lane & 16) ? 8 : 0;
      const int cb  = lane & 15;
      float s0 = 0.f, s1 = 0.f;
#pragma unroll
      for (int v = 0; v < 8; ++v) {
        const int j = j0 + off + v;
        if (j != i) { s0 += g0[v]; s1 += g1[v]; }
      }
      s0 += __shfl_xor(s0, 16, 32);
      s1 += __shfl_xor(s1, 16, 32);
      if (lane < 16) { atomicAdd(&smv[i * 32 + cb], s0); atomicAdd(&smv[i * 32 + 16 + cb], s1); }
    }
  }
  __syncthreads();

  // ---- Phase C: node update (WMMA), waves 0-1 ----
  if (wave < 2) {
    const int m0 = wave << 4;
    // t1 = silu([h_v, m_v] @ Wn1^T + bn1), K=64
    v8f c0 = {}, c1 = {};
    {
      const int row  = lane & 15;
      const int koff = (lane >> 4) << 1;
      const int ncol = lane & 15;
#pragma unroll
      for (int k0 = 0; k0 < 64; k0 += 4) {
        const int ka = k0 + koff, kb = ka + 1;
        v2f a;
        a.x = (ka < 32) ? shv[(m0 + row) * 32 + ka] : smv[(m0 + row) * 32 + ka - 32];
        a.y = (kb < 32) ? shv[(m0 + row) * 32 + kb] : smv[(m0 + row) * 32 + kb - 32];
        v2f b0; b0.x = Wn1[ncol * 64 + ka];        b0.y = Wn1[ncol * 64 + kb];
        v2f b1; b1.x = Wn1[(16 + ncol) * 64 + ka]; b1.y = Wn1[(16 + ncol) * 64 + kb];
        c0 = wmma4(a, b0, c0);
        c1 = wmma4(a, b1, c1);
      }
    }
    c0 = addb_silu(c0, bn1, 0, lane);
    c1 = addb_silu(c1, bn1, 16, lane);
    store_frag(st1 + m0 * 32, c0, 0, lane);
    store_frag(st1 + m0 * 32, c1, 16, lane);
  }
  __syncthreads();
  if (wave < 2) {
    const int m0 = wave << 4;
    // h_v_new = h_v + t1 @ Wn2^T + bn2 ; reduce over nodes into shvSum
    v8f d0 = {}, d1 = {};
    gemm_tile(st1 + m0 * 32, 32, Wn2, 32, d0, d1, lane);
    const int off = (lane & 16) ? 8 : 0;
    const int cb  = lane & 15;
    float s0 = 0.f, s1 = 0.f;
#pragma unroll
    for (int v = 0; v < 8; ++v) {
      const int row = m0 + off + v;
      s0 += d0[v] + bn2[cb]      + shv[row * 32 + cb];
      s1 += d1[v] + bn2[16 + cb] + shv[row * 32 + 16 + cb];
    }
    s0 += __shfl_xor(s0, 16, 32);
    s1 += __shfl_xor(s1, 16, 32);
    if (lane < 16) { atomicAdd(&shvSum[cb], s0); atomicAdd(&shvSum[16 + cb], s1); }
  }
  __syncthreads();

  // ---- Phase D: feature assembly + final MLP + cusp ----
  for (int t = tid; t < 130; t += 128) {
    float v;
    if      (t < 32)  v = shvSum[t];
    else if (t < 64)  v = shvSum[t - 32] * (1.f / 32.f);
    else if (t < 96)  v = sheSum[t - 64];
    else if (t < 128) v = sheSum[t - 96] * (1.f / 496.f);
    else if (t == 128) v = sstat[0] * (1.f / 96.f);
    else               v = sstat[1] * (1.f / 496.f);
    sfin[t] = v;
  }
  __syncthreads();
  if (tid < 64) {
    float a = bf1[tid];
    for (int k = 0; k < 130; ++k) a += Wf1[tid * 130 + k] * sfin[k];
    sl1[tid] = siluf(a);
  }
  __syncthreads();
  if (tid < 64) {
    float a = bf2[tid];
    for (int k = 0; k < 64; ++k) a += Wf2[tid * 64 + k] * sl1[k];
    sl2[tid] = siluf(a);
  }
  __syncthreads();
  if (tid == 0) {
    float a = bf3[0];
    for (int k = 0; k < 64; ++k) a += Wf3[k] * sl2[k];
    out[b] = a + sstat[2];
  }
}

extern "C" void kernel_launch(void* const* d_in, const int* in_sizes, int n_in,
                              void* d_out, int out_size, void* d_ws, size_t ws_size,
                              hipStream_t stream) {
  (void)in_sizes; (void)n_in; (void)out_size; (void)d_ws; (void)ws_size;
  const float* x      = (const float*)d_in[0];
  const float* W_node = (const float*)d_in[1];
  const float* b_node = (const float*)d_in[2];
  const float* We1    = (const float*)d_in[3];
  const float* be1    = (const float*)d_in[4];
  const float* We2    = (const float*)d_in[5];
  const float* be2    = (const float*)d_in[6];
  const float* Wv2e   = (const float*)d_in[7];
  const float* We2v   = (const float*)d_in[8];
  const float* Wu1    = (const float*)d_in[9];
  const float* bu1    = (const float*)d_in[10];
  const float* Wu2    = (const float*)d_in[11];
  const float* bu2    = (const float*)d_in[12];
  const float* Wn1    = (const float*)d_in[13];
  const float* bn1    = (const float*)d_in[14];
  const float* Wn2    = (const float*)d_in[15];
  const float* bn2    = (const float*)d_in[16];
  const float* Wf1    = (const float*)d_in[17];
  const float* bf1    = (const float*)d_in[18];
  const float* Wf2    = (const float*)d_in[19];
  const float* bf2    = (const float*)d_in[20];
  const float* Wf3    = (const float*)d_in[21];
  const float* bf3    = (const float*)d_in[22];
  float* out = (float*)d_out;

  hipLaunchKernelGGL(ctnn_jastrow_kernel, dim3(NB), dim3(128), 0, stream,
                     x, W_node, b_node, We1, be1, We2, be2, Wv2e, We2v,
                     Wu1, bu1, Wu2, bu2, Wn1, bn1, Wn2, bn2,
                     Wf1, bf1, Wf2, bf2, Wf3, bf3, out);
}